// TrustRegionICP_20744692040053
// MI455X (gfx1250) — compile-verified
//
#include <hip/hip_runtime.h>
#include <math.h>

typedef __attribute__((ext_vector_type(2))) float v2f;
typedef __attribute__((ext_vector_type(8))) float v8f;

#define Wd   640
#define Hd   480
#define HWd  (Wd * Hd)          // 307200
#define Bd   8
#define FOCALc    525.0f
#define SIGMA_XYc 5.5f
#define SIGMA_DPc 0.4f

// ---------------------------------------------------------------------------
// init: copy R,t into workspace, zero G accumulators, init min/max sentinels
// ---------------------------------------------------------------------------
__global__ void icp_init(const float* __restrict__ Rin, const float* __restrict__ tin,
                         float* __restrict__ Rw, float* __restrict__ tw,
                         float* __restrict__ G, unsigned* __restrict__ mm) {
  const int tid = threadIdx.x;               // 256 threads
  if (tid < 72) Rw[tid] = Rin[tid];
  if (tid < 24) tw[tid] = tin[tid];
  G[tid] = 0.0f;
  G[256 + tid] = 0.0f;                       // 8 batches * 64 = 512 entries
  if (tid == 0) { mm[0] = 0x7F800000u; mm[1] = 0u; }
}

// ---------------------------------------------------------------------------
// global min/max of depth1 (all positive -> uint bit ordering == float order)
// ---------------------------------------------------------------------------
__global__ __launch_bounds__(256) void icp_minmax(const float* __restrict__ depth1,
                                                  unsigned* __restrict__ mm) {
  __shared__ float smn[256], smx[256];
  const int tid = threadIdx.x;
  float mn = 3.0e38f, mx = -3.0e38f;
  for (int i = blockIdx.x * 256 + tid; i < Bd * HWd; i += gridDim.x * 256) {
    const float d = depth1[i];
    mn = fminf(mn, d);
    mx = fmaxf(mx, d);
  }
  smn[tid] = mn; smx[tid] = mx;
  __syncthreads();
  for (int s = 128; s > 0; s >>= 1) {
    if (tid < s) {
      smn[tid] = fminf(smn[tid], smn[tid + s]);
      smx[tid] = fmaxf(smx[tid], smx[tid + s]);
    }
    __syncthreads();
  }
  if (tid == 0) {
    atomicMin(&mm[0], __float_as_uint(smn[0]));
    atomicMax(&mm[1], __float_as_uint(smx[0]));
  }
}

// ---------------------------------------------------------------------------
// normals of vertex1 via Sobel (edge-padded), normalized, invalid mask -> 0
// stored as float4 (nx,ny,nz,0) so the warp gather is one b128 per corner
// ---------------------------------------------------------------------------
__global__ __launch_bounds__(256) void icp_normal(const float* __restrict__ depth1,
                                                  const float* __restrict__ Km,
                                                  const unsigned* __restrict__ mm,
                                                  float4* __restrict__ nrm) {
  const int idx = blockIdx.x * 256 + threadIdx.x;
  if (idx >= Bd * HWd) return;
  const int b = idx / HWd;
  const int p = idx - b * HWd;
  const int u = p % Wd;
  const int v = p / Wd;
  const float fx = Km[b * 4 + 0], fy = Km[b * 4 + 1];
  const float cx = Km[b * 4 + 2], cy = Km[b * 4 + 3];
  const float SXk[3][3] = {{-1.f, 0.f, 1.f}, {-2.f, 0.f, 2.f}, {-1.f, 0.f, 1.f}};
  const float SYk[3][3] = {{-1.f, -2.f, -1.f}, {0.f, 0.f, 0.f}, {1.f, 2.f, 1.f}};
  const float* dp = depth1 + b * HWd;
  float gx0 = 0.f, gx1 = 0.f, gx2 = 0.f, gy0 = 0.f, gy1 = 0.f, gy2 = 0.f;
#pragma unroll
  for (int i = 0; i < 3; ++i) {
    const int vv = min(max(v - 1 + i, 0), Hd - 1);
#pragma unroll
    for (int j = 0; j < 3; ++j) {
      const int uu = min(max(u - 1 + j, 0), Wd - 1);
      const float d = dp[vv * Wd + uu];
      const float vx = ((float)uu - cx) / fx * d;
      const float vy = ((float)vv - cy) / fy * d;
      const float sx = SXk[i][j], sy = SYk[i][j];
      gx0 += sx * vx; gx1 += sx * vy; gx2 += sx * d;
      gy0 += sy * vx; gy1 += sy * vy; gy2 += sy * d;
    }
  }
  float nx = gx1 * gy2 - gx2 * gy1;
  float ny = gx2 * gy0 - gx0 * gy2;
  float nz = gx0 * gy1 - gx1 * gy0;
  const float mag = sqrtf(nx * nx + ny * ny + nz * nz);
  const float inv = 1.0f / (mag + 1e-8f);
  nx *= inv; ny *= inv; nz *= inv;
  const float dc = dp[p];
  const float dmin = __uint_as_float(mm[0]);
  const float dmax = __uint_as_float(mm[1]);
  if (dc == dmin || dc == dmax) { nx = 0.f; ny = 0.f; nz = 0.f; }
  nrm[idx] = make_float4(nx, ny, nz, 0.f);
}

// ---------------------------------------------------------------------------
// weights output = ones
// ---------------------------------------------------------------------------
__global__ void icp_fill_ones(float* __restrict__ w) {
  w[blockIdx.x * 256 + threadIdx.x] = 1.0f;
}

// ---------------------------------------------------------------------------
// main reduction: per point build q = [-J/sigma (6), res'(1)], accumulate
// G = sum q q^T per batch with V_WMMA_F32_16X16X4_F32 (A operand == B operand
// by the ISA 32-bit A/B VGPR layouts). grid = (300 chunks, 8 batches),
// 256 threads (8 waves), chunk = 1024 points (exactly covers 307200).
// ---------------------------------------------------------------------------
__global__ __launch_bounds__(256) void icp_reduce(const float* __restrict__ depth0,
                                                  const float* __restrict__ depth1,
                                                  const float* __restrict__ Km,
                                                  const float* __restrict__ Rw,
                                                  const float* __restrict__ tw,
                                                  const float4* __restrict__ nrm,
                                                  float* __restrict__ G) {
  __shared__ float sq[256 * 16];   // 16 KB staging: 256 points x 16 padded cols
  __shared__ float Gs[64];         // block-level 8x8 accumulator
  const int tid  = threadIdx.x;
  const int lane = tid & 31;
  const int wave = tid >> 5;
  const int b    = blockIdx.y;
  const int base = blockIdx.x * 1024;

  // zero padded columns (8..15 never touched again) and block accumulator
#pragma unroll
  for (int c = 8; c < 16; ++c) sq[tid * 16 + c] = 0.0f;
  if (tid < 64) Gs[tid] = 0.0f;

  const float fx = Km[b * 4 + 0], fy = Km[b * 4 + 1];
  const float cx = Km[b * 4 + 2], cy = Km[b * 4 + 3];
  float Rm[9];
#pragma unroll
  for (int i = 0; i < 9; ++i) Rm[i] = Rw[b * 9 + i];
  const float tx = tw[b * 3 + 0], ty = tw[b * 3 + 1], tz = tw[b * 3 + 2];

  const float*  d0p = depth0 + b * HWd;
  const float*  d1p = depth1 + b * HWd;
  const float4* npz = nrm + b * HWd;

  v8f acc = {0.f, 0.f, 0.f, 0.f, 0.f, 0.f, 0.f, 0.f};
  __syncthreads();

#pragma unroll 1
  for (int it = 0; it < 4; ++it) {
    const int p = base + it * 256 + tid;     // always < HWd (300*1024 == HWd)
    const int u = p % Wd;
    const int v = p / Wd;

    const float d0  = d0p[p];
    const float v0x = ((float)u - cx) / fx * d0;
    const float v0y = ((float)v - cy) / fy * d0;
    const float v0z = d0;
    // v01 = R v0 + t
    const float wxv = Rm[0] * v0x + Rm[1] * v0y + Rm[2] * v0z + tx;
    const float wyv = Rm[3] * v0x + Rm[4] * v0y + Rm[5] * v0z + ty;
    const float wzv = Rm[6] * v0x + Rm[7] * v0y + Rm[8] * v0z + tz;
    const float iz  = 1.0f / wzv;
    const float uf  = wxv * iz * fx + cx;
    const float vf  = wyv * iz * fy + cy;
    const bool inview = (uf > 0.f) && (uf < (float)(Wd - 1)) &&
                        (vf > 0.f) && (vf < (float)(Hd - 1));
    const float ucl = fminf(fmaxf(uf, 0.f), (float)(Wd - 1));
    const float vcl = fminf(fmaxf(vf, 0.f), (float)(Hd - 1));
    const float u0f = floorf(ucl), v0f = floorf(vcl);
    const float fu = ucl - u0f, fv = vcl - v0f;
    const int u0i = (int)u0f, v0i = (int)v0f;
    const int u1i = min(u0i + 1, Wd - 1), v1i = min(v0i + 1, Hd - 1);
    const float w00 = (1.f - fu) * (1.f - fv), w01 = fu * (1.f - fv);
    const float w10 = (1.f - fu) * fv,         w11 = fu * fv;
    const int i00 = v0i * Wd + u0i, i01 = v0i * Wd + u1i;
    const int i10 = v1i * Wd + u0i, i11 = v1i * Wd + u1i;

    float rvx = 0.f, rvy = 0.f, rvz = 0.f, nx = 0.f, ny = 0.f, nz = 0.f;
    {
      const int   ci[4] = {i00, i01, i10, i11};
      const int   cu[4] = {u0i, u1i, u0i, u1i};
      const int   cv[4] = {v0i, v0i, v1i, v1i};
      const float cw[4] = {w00, w01, w10, w11};
#pragma unroll
      for (int c = 0; c < 4; ++c) {
        const float d1 = d1p[ci[c]];
        const float pxc = ((float)cu[c] - cx) / fx;
        const float pyc = ((float)cv[c] - cy) / fy;
        rvx += cw[c] * pxc * d1;
        rvy += cw[c] * pyc * d1;
        rvz += cw[c] * d1;
        const float4 nn = npz[ci[c]];
        nx += cw[c] * nn.x; ny += cw[c] * nn.y; nz += cw[c] * nn.z;
      }
    }

    const float dxv = wxv - rvx, dyv = wyv - rvy, dzv = wzv - rvz;
    const float res0 = nx * dxv + ny * dyv + nz * dzv;
    const bool occ = (!inview) || (dxv * dxv + dyv * dyv + dzv * dzv > 0.01f);
    // Jrot = -(n^T skew(v01)) = v01 x n
    const float jr0 = wyv * nz - wzv * ny;
    const float jr1 = wzv * nx - wxv * nz;
    const float jr2 = wxv * ny - wyv * nx;
    const float sdxy = d0 * (SIGMA_XYc / FOCALc);
    const float sdz  = d0 * d0 * (SIGMA_DPc / FOCALc);
    const float jn0 = nx * Rm[0] + ny * Rm[3] + nz * Rm[6];
    const float jn1 = nx * Rm[1] + ny * Rm[4] + nz * Rm[7];
    const float jn2 = nx * Rm[2] + ny * Rm[5] + nz * Rm[8];
    const float sigma = sqrtf((jn0 * jn0 + jn1 * jn1) * sdxy * sdxy +
                              jn2 * jn2 * sdz * sdz + 1e-8f);
    const float isg = 1.0f / (sigma + 1e-8f);

    float* row = &sq[tid * 16];
    row[0] = -jr0 * isg;
    row[1] = -jr1 * isg;
    row[2] = -jr2 * isg;
    row[3] = -nx * isg;
    row[4] = -ny * isg;
    row[5] = -nz * isg;
    row[6] = occ ? 1e-6f : res0 * isg;   // only residual is masked (matches ref)
    row[7] = 0.0f;
    __syncthreads();

    // 8 WMMAs consume this wave's 32 staged points, K=4 points each.
    const int m       = lane & 15;
    const int halfsel = (lane >> 4) << 1;       // 0 for lanes 0-15, 2 for 16-31
#pragma unroll
    for (int s = 0; s < 8; ++s) {
      const int r0 = wave * 32 + s * 4 + halfsel;
      v2f a;
      a.x = sq[r0 * 16 + m];
      a.y = sq[(r0 + 1) * 16 + m];
      // D = A * B + C with A == B (G is symmetric: D[m][n] += sum_k q_k[m] q_k[n])
      acc = __builtin_amdgcn_wmma_f32_16x16x4_f32(
          false, a, false, a, (short)0, acc, false, false);
    }
    __syncthreads();
  }

  // D layout: VGPR r, lanes 0-15 hold D[M=r][N=lane]; only 7x7 corner nonzero.
  if (lane < 7) {
#pragma unroll
    for (int r = 0; r < 7; ++r) atomicAdd(&Gs[r * 8 + lane], acc[r]);
  }
  __syncthreads();
  if (tid < 64) {
    const int r = tid >> 3, c = tid & 7;
    if (r < 7 && c < 7) atomicAdd(&G[b * 64 + tid], Gs[tid]);
  }
}

// ---------------------------------------------------------------------------
// per-batch 6x6 damped solve + SE3 update; zeroes G for next iteration;
// optionally writes final R,t to output (last iteration).
// ---------------------------------------------------------------------------
__global__ void icp_solve(float* __restrict__ Rw, float* __restrict__ tw,
                          float* __restrict__ G, float* __restrict__ out,
                          int write_out) {
  const int b = blockIdx.x;
  if (threadIdx.x != 0) return;

  float A[6][7];
  float trace = 0.f;
  for (int i = 0; i < 6; ++i) {
    for (int j = 0; j < 6; ++j) A[i][j] = G[b * 64 + i * 8 + j];
    A[i][6] = G[b * 64 + i * 8 + 6];       // JtR
    trace += A[i][i];
  }
  const float damp = trace * 1e-6f;
  for (int i = 0; i < 6; ++i) A[i][i] += damp;

  // Gauss-Jordan with partial pivoting: solve (JtJ + damp I) xi = JtR
  for (int col = 0; col < 6; ++col) {
    int piv = col;
    for (int r = col + 1; r < 6; ++r)
      if (fabsf(A[r][col]) > fabsf(A[piv][col])) piv = r;
    if (piv != col)
      for (int j = 0; j < 7; ++j) { float tmp = A[col][j]; A[col][j] = A[piv][j]; A[piv][j] = tmp; }
    const float ip = 1.0f / A[col][col];
    for (int j = 0; j < 7; ++j) A[col][j] *= ip;
    for (int r = 0; r < 6; ++r) {
      if (r == col) continue;
      const float f = A[r][col];
      for (int j = 0; j < 7; ++j) A[r][j] -= f * A[col][j];
    }
  }
  float xi[6];
  for (int i = 0; i < 6; ++i) xi[i] = A[i][6];

  // dR = I + A*W + B*W^2  (Rodrigues, matching _twist2mat epsilons)
  const float w0 = xi[0], w1 = xi[1], w2 = xi[2];
  const float th2 = w0 * w0 + w1 * w1 + w2 * w2;
  const float th = sqrtf(th2 + 1e-12f);
  const float Ac = sinf(th) / th;
  const float Bc = (1.0f - cosf(th)) / (th2 + 1e-12f);
  const float Wm[9] = {0.f, -w2, w1, w2, 0.f, -w0, -w1, w0, 0.f};
  float W2[9];
  for (int i = 0; i < 3; ++i)
    for (int j = 0; j < 3; ++j) {
      float s = 0.f;
      for (int k = 0; k < 3; ++k) s += Wm[i * 3 + k] * Wm[k * 3 + j];
      W2[i * 3 + j] = s;
    }
  float dR[9];
  for (int i = 0; i < 9; ++i)
    dR[i] = ((i % 4 == 0) ? 1.0f : 0.0f) + Ac * Wm[i] + Bc * W2[i];

  float Ro[9], to[3];
  for (int i = 0; i < 9; ++i) Ro[i] = Rw[b * 9 + i];
  for (int i = 0; i < 3; ++i) to[i] = tw[b * 3 + i];
  float Rn[9], tn[3];
  for (int i = 0; i < 3; ++i) {
    for (int j = 0; j < 3; ++j) {
      float s = 0.f;
      for (int k = 0; k < 3; ++k) s += dR[i * 3 + k] * Ro[k * 3 + j];
      Rn[i * 3 + j] = s;
    }
    tn[i] = dR[i * 3 + 0] * to[0] + dR[i * 3 + 1] * to[1] + dR[i * 3 + 2] * to[2] + xi[3 + i];
  }
  for (int i = 0; i < 9; ++i) Rw[b * 9 + i] = Rn[i];
  for (int i = 0; i < 3; ++i) tw[b * 3 + i] = tn[i];
  for (int i = 0; i < 64; ++i) G[b * 64 + i] = 0.0f;   // ready for next iter
  if (write_out) {
    for (int i = 0; i < 9; ++i) out[b * 9 + i] = Rn[i];
    for (int i = 0; i < 3; ++i) out[72 + b * 3 + i] = tn[i];
  }
}

// ---------------------------------------------------------------------------
extern "C" void kernel_launch(void* const* d_in, const int* in_sizes, int n_in,
                              void* d_out, int out_size, void* d_ws, size_t ws_size,
                              hipStream_t stream) {
  (void)in_sizes; (void)n_in; (void)out_size; (void)ws_size;
  const float* Rin    = (const float*)d_in[0];   // 8*3*3
  const float* tin    = (const float*)d_in[1];   // 8*3
  const float* depth0 = (const float*)d_in[2];   // 8*1*480*640
  const float* depth1 = (const float*)d_in[3];   // 8*1*480*640
  const float* Km     = (const float*)d_in[4];   // 8*4
  float* out = (float*)d_out;                    // [R(72) | t(24) | weights(2457600)]

  float*    ws  = (float*)d_ws;
  float*    Rw  = ws;                            // 72
  float*    tw  = ws + 72;                       // 24
  float*    G   = ws + 96;                       // 8 * 64
  unsigned* mm  = (unsigned*)(ws + 608);         // min/max bits
  float4*   nrm = (float4*)(ws + 640);           // 8*HW float4 (39.3 MB)

  icp_init<<<1, 256, 0, stream>>>(Rin, tin, Rw, tw, G, mm);
  icp_minmax<<<1200, 256, 0, stream>>>(depth1, mm);
  icp_normal<<<(Bd * HWd) / 256, 256, 0, stream>>>(depth1, Km, mm, nrm);
  icp_fill_ones<<<(Bd * HWd) / 256, 256, 0, stream>>>(out + 96);

  for (int it = 0; it < 3; ++it) {
    icp_reduce<<<dim3(300, Bd), 256, 0, stream>>>(depth0, depth1, Km, Rw, tw, nrm, G);
    icp_solve<<<Bd, 32, 0, stream>>>(Rw, tw, G, out, (it == 2) ? 1 : 0);
  }
}